// Factor1d_70755291234415
// MI455X (gfx1250) — compile-verified
//
#include <hip/hip_runtime.h>

typedef __attribute__((ext_vector_type(2))) float v2f;
typedef __attribute__((ext_vector_type(8))) float v8f;

#define B_   32
#define N_   50000
#define F_   50000
#define K_   32
#define H_   10
#define EPS_ 1e-5f
#define FPB  8   // features per block (one per wave32)

__global__ __launch_bounds__(256) void factor1d_wmma_kernel(
    const float* __restrict__ x,      // (B, 1, N)
    const int*   __restrict__ idx,    // (F, K)
    const float* __restrict__ w0,     // (F, H, 1, K)
    const float* __restrict__ gamma,  // (F, H)
    const float* __restrict__ beta,   // (F, H)
    const float* __restrict__ w1,     // (F, 1, H)
    float*       __restrict__ out)    // (B, 1, F)
{
    __shared__ float lds_out[B_ * FPB];

    const int lane = threadIdx.x & 31;
    const int wave = threadIdx.x >> 5;
    const int f    = blockIdx.x * FPB + wave;

    const int hi = lane >> 4;   // which half of the wave
    const int lo = lane & 15;

    // Each lane owns one idx element of this feature's gather row.
    const int idxv = idx[f * K_ + lane];

    // Prefetch this feature's 1.25KB weight row into cache (global_prefetch_b8).
    __builtin_prefetch(&w0[(size_t)f * (H_ * K_)], 0, 0);

    // Column h of the B / C matrices handled by this lane (N dim padded 10->16).
    const int   h  = lo;
    const int   hc = (h < H_) ? h : (H_ - 1);   // clamped (in-bounds) index
    const float hv = (h < H_) ? 1.0f : 0.0f;    // zero-pad mask

    const float* w0row = w0 + (size_t)f * (H_ * K_) + hc * K_;
    const float* xr0   = x + (size_t)lo * N_;          // batch rows 0..15
    const float* xr1   = x + (size_t)(lo + 16) * N_;   // batch rows 16..31

    v8f c0 = {};   // C tile, rows b = 0..15
    v8f c1 = {};   // C tile, rows b = 16..31

#pragma unroll
    for (int kk = 0; kk < 8; ++kk) {
        // A-layout (f32 16x16x4): lo lanes own K = 4kk,4kk+1 ; hi lanes K = 4kk+2,4kk+3
        const int k0 = 4 * kk + 2 * hi;
        const int k1 = k0 + 1;
        const int i0 = __shfl(idxv, k0, 32);
        const int i1 = __shfl(idxv, k1, 32);

        v2f a0, a1, bt;
        a0.x = xr0[i0];  a0.y = xr0[i1];            // gathered G rows 0..15
        a1.x = xr1[i0];  a1.y = xr1[i1];            // gathered G rows 16..31
        bt.x = w0row[k0] * hv;                      // B[k][h] = w0[f,h,k], padded
        bt.y = w0row[k1] * hv;

        c0 = __builtin_amdgcn_wmma_f32_16x16x4_f32(false, a0, false, bt,
                                                   (short)0, c0, false, false);
        c1 = __builtin_amdgcn_wmma_f32_16x16x4_f32(false, a1, false, bt,
                                                   (short)0, c1, false, false);
    }

    // ---- BatchNorm over the 32 batch rows, per column h ----
    // Column h lives in lanes {h, h+16}; each holds 8 rows per C tile.
    float s = 0.f, q = 0.f;
#pragma unroll
    for (int r = 0; r < 8; ++r) {
        s += c0[r] + c1[r];
        q += c0[r] * c0[r] + c1[r] * c1[r];
    }
    s += __shfl_xor(s, 16, 32);
    q += __shfl_xor(q, 16, 32);
    const float mean = s * (1.0f / 32.0f);
    const float var  = q * (1.0f / 32.0f) - mean * mean;   // population var (ddof=0)
    const float rs   = rsqrtf(var + EPS_);

    const float g   = gamma[f * H_ + hc];
    const float be  = beta [f * H_ + hc];
    const float w1v = w1   [f * H_ + hc] * hv;             // zero for padded columns

    // ---- normalize + ReLU + weight by w1, reduce over h (16-lane halves) ----
#pragma unroll
    for (int r = 0; r < 8; ++r) {
        float t0 = (c0[r] - mean) * rs * g + be;
        float t1 = (c1[r] - mean) * rs * g + be;
        t0 = fmaxf(t0, 0.f) * w1v;
        t1 = fmaxf(t1, 0.f) * w1v;
        t0 += __shfl_xor(t0, 1, 32);
        t0 += __shfl_xor(t0, 2, 32);
        t0 += __shfl_xor(t0, 4, 32);
        t0 += __shfl_xor(t0, 8, 32);   // all 16 lanes of each half now hold Σ_h
        t1 += __shfl_xor(t1, 1, 32);
        t1 += __shfl_xor(t1, 2, 32);
        t1 += __shfl_xor(t1, 4, 32);
        t1 += __shfl_xor(t1, 8, 32);
        if (lo == r) {
            const int row = r + 8 * hi;                    // C layout: hi half = rows +8
            lds_out[row * FPB + wave]        = t0;         // rows 0..15
            lds_out[(row + 16) * FPB + wave] = t1;         // rows 16..31
        }
    }

    __syncthreads();

    // Store: thread t -> (b = t/8, w = t%8); 8 consecutive f per batch row.
    const int b  = threadIdx.x >> 3;
    const int w  = threadIdx.x & 7;
    const int fo = blockIdx.x * FPB + w;
    out[(size_t)b * F_ + fo] = lds_out[b * FPB + w];
}

extern "C" void kernel_launch(void* const* d_in, const int* in_sizes, int n_in,
                              void* d_out, int out_size, void* d_ws, size_t ws_size,
                              hipStream_t stream) {
    const float* x     = (const float*)d_in[0];
    const int*   idx   = (const int*)  d_in[1];
    const float* w0    = (const float*)d_in[2];
    const float* gamma = (const float*)d_in[3];
    const float* beta  = (const float*)d_in[4];
    const float* w1    = (const float*)d_in[5];
    float*       out   = (float*)d_out;

    const int blocks = F_ / FPB;   // 6250
    factor1d_wmma_kernel<<<blocks, 256, 0, stream>>>(x, idx, w0, gamma, beta, w1, out);
}